// EdgeActivation_77206332113791
// MI455X (gfx1250) — compile-verified
//
#include <hip/hip_runtime.h>
#include <hip/hip_bf16.h>
#include <cstdint>

#ifndef __has_builtin
#define __has_builtin(x) 0
#endif

#define TPB 256

static constexpr float kTenPi = 31.415926535897932f;

// Native 4x f32 vector (works with __builtin_nontemporal_store, lowers to b128).
typedef float v4f __attribute__((ext_vector_type(4)));

// ---------------------------------------------------------------------------
// CDNA5 async global->LDS copy (ASYNCcnt path, CDNA5 ISA).
// GV mode: 64-bit per-lane vaddr, per-lane LDS dest offset in VDST.
// TH_LOAD_NT: input is streamed once, keep it out of WGP$/L2 (cache doc 7.3).
// ---------------------------------------------------------------------------
__device__ __forceinline__ void async_load_f4_to_lds(const float* gsrc, float* ldst) {
    // Low 32 bits of a generic pointer into LDS are the LDS byte offset
    // (ISA 10.2).
    unsigned lds_off = (unsigned)(unsigned long long)(uintptr_t)ldst;
    unsigned long long ga = (unsigned long long)(uintptr_t)gsrc;
    asm volatile("global_load_async_to_lds_b128 %0, %1, off th:TH_LOAD_NT"
                 :: "v"(lds_off), "v"(ga)
                 : "memory");
}

__device__ __forceinline__ void wait_async_le1() {
    asm volatile("s_wait_asynccnt 0x1" ::: "memory");
}
__device__ __forceinline__ void wait_async_le0() {
    asm volatile("s_wait_asynccnt 0x0" ::: "memory");
}

// ---------------------------------------------------------------------------
// The 9-basis weighted activation. Native TRANS32 ops: exp/log/sqrt/sin/tanh.
// ---------------------------------------------------------------------------
__device__ __forceinline__ float edge_act(float x, const float* __restrict__ w, float b) {
    float ax = __builtin_fabsf(x);
    float x2 = fminf(x * x, 100.0f);                                  // clip(x^2)
    float x3 = fminf(fmaxf(x * x * x, -100.0f), 100.0f);              // clip(x^3)
    float ex = __expf(fminf(fmaxf(x, -10.0f), 10.0f));                // exp(clip)
    float lg = __logf(ax + 1.0f);                                     // log(|x|+1)
    float sq = __builtin_amdgcn_sqrtf(ax);                            // sqrt(|x|)
#if __has_builtin(__builtin_amdgcn_tanhf)
    float th = __builtin_amdgcn_tanhf(x);                             // v_tanh_f32
#else
    float e2 = __expf(-2.0f * ax);
    float th = __fdividef(1.0f - e2, 1.0f + e2);
    th = __builtin_copysignf(th, x);
#endif
    float sn = __sinf(fminf(fmaxf(x, -kTenPi), kTenPi));              // sin(clip)

    float acc = fmaf(w[0], x,  b);
    acc = fmaf(w[1], x2, acc);
    acc = fmaf(w[2], x3, acc);
    acc = fmaf(w[3], ex, acc);
    acc = fmaf(w[4], lg, acc);
    acc = fmaf(w[5], sq, acc);
    acc = fmaf(w[6], th, acc);
    acc = fmaf(w[7], sn, acc);
    acc = fmaf(w[8], ax, acc);
    return acc;
}

// ---------------------------------------------------------------------------
// Streaming kernel: async double-buffered global->LDS->VGPR pipeline.
// Each thread owns one float4 slot per buffer; no cross-thread sharing, so no
// barriers are needed -- only s_wait_asynccnt between stages.
// ---------------------------------------------------------------------------
__global__ __launch_bounds__(TPB) void edge_activation_kernel(
    const float* __restrict__ x,
    const float* __restrict__ w9,
    const float* __restrict__ bias,
    float* __restrict__ out,
    long long n)
{
    __shared__ __align__(16) float buf[2][TPB * 4];

    const int tid = threadIdx.x;
    const long long n4 = n >> 2;
    const long long stride = (long long)gridDim.x * TPB;
    long long idx = (long long)blockIdx.x * TPB + tid;

    float w[9];
#pragma unroll
    for (int k = 0; k < 9; ++k) w[k] = w9[k];
    const float b = bias[0];

    float* lds0 = &buf[0][tid * 4];
    float* lds1 = &buf[1][tid * 4];

    if (idx < n4) async_load_f4_to_lds(x + 4 * idx, lds0);

    int s = 0;
    for (long long i = idx; i < n4; i += stride, s ^= 1) {
        const long long nxt = i + stride;
        const bool have_next = nxt < n4;
        if (have_next) {
            async_load_f4_to_lds(x + 4 * nxt, s ? lds0 : lds1);
            wait_async_le1();   // stage i has landed in LDS
        } else {
            wait_async_le0();
        }

        const v4f v = *reinterpret_cast<const v4f*>(s ? lds1 : lds0);

        v4f o;
        o.x = edge_act(v.x, w, b);
        o.y = edge_act(v.y, w, b);
        o.z = edge_act(v.z, w, b);
        o.w = edge_act(v.w, w, b);

        // Streaming output: global_store_b128 with TH=NT (no L2 residency).
        __builtin_nontemporal_store(o, reinterpret_cast<v4f*>(out + 4 * i));
    }

    // Scalar tail (n % 4 != 0); no-op for the 4096x8192 reference shape.
    const long long base = n4 << 2;
    const long long rem = n - base;
    const long long t = (long long)blockIdx.x * TPB + tid;
    if (t < rem) {
        out[base + t] = edge_act(x[base + t], w, b);
    }
}

extern "C" void kernel_launch(void* const* d_in, const int* in_sizes, int n_in,
                              void* d_out, int out_size, void* d_ws, size_t ws_size,
                              hipStream_t stream) {
    (void)in_sizes; (void)n_in; (void)d_ws; (void)ws_size;

    const float* x    = (const float*)d_in[0];   // [4096, 8192] fp32
    const float* w    = (const float*)d_in[1];   // [9] fp32
    const float* bias = (const float*)d_in[2];   // [1] fp32
    float* out        = (float*)d_out;           // [4096, 8192, 1] fp32

    const long long n  = (long long)out_size;    // 33,554,432 elements
    if (n <= 0) return;

    const long long n4 = n >> 2;
    long long need = (n4 + TPB - 1) / TPB;
    if (need < 1) need = 1;
    int blocks = (need < 4096) ? (int)need : 4096;  // 8 pipelined stages/thread at full size

    edge_activation_kernel<<<blocks, TPB, 0, stream>>>(x, w, bias, out, n);
}